// self_loss_1597727834617
// MI455X (gfx1250) — compile-verified
//
#include <hip/hip_runtime.h>
#include <math.h>

#define NUM    5
#define BATCH  32768
#define LABELS 1000
#define CHUNKS 250   // LABELS / 4 float4 chunks per row (4000 B, 16B-aligned)

typedef __attribute__((ext_vector_type(2))) float v2f;
typedef __attribute__((ext_vector_type(8))) float v8f;

// Exact fp32 sum of one value per lane across a wave32.
// Uses V_WMMA_F32_16X16X4_F32: A[m][0]=s_m (lanes 0-15), A[m][2]=s_{m+16}
// (lanes 16-31), K1/K3 zero, B=ones  =>  D[m][n] = s_m + s_{m+16}.
// Per-lane D holds M=lane<16 ? 0..7 : 8..15, so summing the 8 VGPRs gives two
// halves; one shfl_xor(16) combines them.
__device__ __forceinline__ float wave_sum32(float s) {
#if __has_builtin(__builtin_amdgcn_wmma_f32_16x16x4_f32)
  v2f a; a.x = s;    a.y = 0.0f;
  v2f b; b.x = 1.0f; b.y = 1.0f;
  v8f c = {0.f, 0.f, 0.f, 0.f, 0.f, 0.f, 0.f, 0.f};
  v8f d = __builtin_amdgcn_wmma_f32_16x16x4_f32(
      /*neg_a=*/false, a, /*neg_b=*/false, b,
      /*c_mod=*/(short)0, c, /*reuse_a=*/false, /*reuse_b=*/false);
  float t = ((d[0] + d[1]) + (d[2] + d[3])) + ((d[4] + d[5]) + (d[6] + d[7]));
  t += __shfl_xor(t, 16, 32);
  return t;
#else
  #pragma unroll
  for (int off = 16; off; off >>= 1) s += __shfl_xor(s, off, 32);
  return s;
#endif
}

// Kernel 1: one wave per row. Single HBM pass: buffer row in VGPRs,
// compute first-occurrence argmax + stable log-sum-exp.
__global__ __launch_bounds__(256) void rowstats_kernel(
    const float* __restrict__ logits, int* __restrict__ labs,
    float* __restrict__ logz) {
  const int lane = threadIdx.x & 31;
  const int wib  = threadIdx.x >> 5;
  const size_t row = (size_t)blockIdx.x * 8 + wib;  // [0, NUM*BATCH)
  const float4* __restrict__ rp =
      (const float4*)(logits + row * (size_t)LABELS);

  float4 v[8];
  #pragma unroll
  for (int k = 0; k < 8; ++k) {
    const int chunk = k * 32 + lane;
    if (chunk < CHUNKS) {
      v[k] = rp[chunk];
    } else {
      v[k] = make_float4(-__builtin_inff(), -__builtin_inff(),
                         -__builtin_inff(), -__builtin_inff());
    }
  }

  // Lane-local argmax; indices strictly increase within a lane, so strict '>'
  // preserves first-occurrence semantics.
  float m = -__builtin_inff();
  int   mi = 0;
  #pragma unroll
  for (int k = 0; k < 8; ++k) {
    const int base = (k * 32 + lane) * 4;
    float x;
    x = v[k].x; if (x > m) { m = x; mi = base + 0; }
    x = v[k].y; if (x > m) { m = x; mi = base + 1; }
    x = v[k].z; if (x > m) { m = x; mi = base + 2; }
    x = v[k].w; if (x > m) { m = x; mi = base + 3; }
  }
  // Cross-lane argmax, tie -> smaller index (matches jnp.argmax).
  #pragma unroll
  for (int off = 16; off; off >>= 1) {
    float om = __shfl_xor(m, off, 32);
    int   oi = __shfl_xor(mi, off, 32);
    if (om > m || (om == m && oi < mi)) { m = om; mi = oi; }
  }

  // Sum of exp(x - max) over the row (tail chunks masked out).
  float s = 0.0f;
  #pragma unroll
  for (int k = 0; k < 8; ++k) {
    const int chunk = k * 32 + lane;
    if (chunk < CHUNKS) {
      s += __expf(v[k].x - m);
      s += __expf(v[k].y - m);
      s += __expf(v[k].z - m);
      s += __expf(v[k].w - m);
    }
  }
  const float total = wave_sum32(s);

  if (lane == 0) {
    labs[row] = mi;
    logz[row] = m + __logf(total);
  }
}

// Kernel 2: exact reference vote tie-break + gather picked logits,
// block-reduce to one partial per block.
__global__ __launch_bounds__(256) void vote_pick_kernel(
    const float* __restrict__ logits, const int* __restrict__ labs,
    const float* __restrict__ logz, float* __restrict__ partials) {
  const int b = blockIdx.x * 256 + threadIdx.x;  // [0, BATCH)

  int lab[NUM];
  #pragma unroll
  for (int j = 0; j < NUM; ++j) lab[j] = labs[j * BATCH + b];

  // Reference: counts[lab]++ ; c = counts[lab]; if (c > best_count) update.
  int bestc = 0, bestl = 0;
  #pragma unroll
  for (int j = 0; j < NUM; ++j) {
    int c = 1;
    #pragma unroll
    for (int i = 0; i < NUM; ++i)
      if (i < j && lab[i] == lab[j]) ++c;
    if (c > bestc) { bestc = c; bestl = lab[j]; }
  }

  float acc = 0.0f;
  #pragma unroll
  for (int j = 0; j < NUM; ++j) {
    const size_t idx =
        ((size_t)j * BATCH + (size_t)b) * (size_t)LABELS + (size_t)bestl;
    acc += logits[idx] - logz[j * BATCH + b];
  }

  #pragma unroll
  for (int off = 16; off; off >>= 1) acc += __shfl_xor(acc, off, 32);
  __shared__ float sm[8];
  if ((threadIdx.x & 31) == 0) sm[threadIdx.x >> 5] = acc;
  __syncthreads();
  if (threadIdx.x == 0) {
    float t = 0.0f;
    #pragma unroll
    for (int w = 0; w < 8; ++w) t += sm[w];
    partials[blockIdx.x] = t;
  }
}

// Kernel 3: deterministic final reduction of 128 partials -> scalar loss.
__global__ __launch_bounds__(128) void final_reduce_kernel(
    const float* __restrict__ partials, float* __restrict__ out) {
  float v = partials[threadIdx.x];  // 128 threads, 128 partials
  #pragma unroll
  for (int off = 16; off; off >>= 1) v += __shfl_xor(v, off, 32);
  __shared__ float sm[4];
  if ((threadIdx.x & 31) == 0) sm[threadIdx.x >> 5] = v;
  __syncthreads();
  if (threadIdx.x == 0) {
    const float t = (sm[0] + sm[1]) + (sm[2] + sm[3]);
    out[0] = -t * (1.0f / (float)BATCH);
  }
}

extern "C" void kernel_launch(void* const* d_in, const int* in_sizes, int n_in,
                              void* d_out, int out_size, void* d_ws,
                              size_t ws_size, hipStream_t stream) {
  (void)in_sizes; (void)n_in; (void)out_size; (void)ws_size;
  const float* logits = (const float*)d_in[0];

  // Workspace layout.
  int*   labs     = (int*)d_ws;                                   // 5*B ints
  float* logz     = (float*)((char*)d_ws + (size_t)NUM * BATCH * 4);
  float* partials = (float*)((char*)d_ws + (size_t)2 * NUM * BATCH * 4);
  float* out      = (float*)d_out;

  // 163840 rows, 8 waves (rows) per 256-thread block -> 20480 blocks.
  rowstats_kernel<<<(NUM * BATCH) / 8, 256, 0, stream>>>(logits, labs, logz);
  // 32768 batch elements -> 128 blocks of 256.
  vote_pick_kernel<<<BATCH / 256, 256, 0, stream>>>(logits, labs, logz,
                                                    partials);
  final_reduce_kernel<<<1, 128, 0, stream>>>(partials, out);
}